// NeuralODE_89824946028939
// MI455X (gfx1250) — compile-verified
//
#include <hip/hip_runtime.h>

#if defined(__HIP_DEVICE_COMPILE__) && !__has_builtin(__builtin_amdgcn_wmma_f32_16x16x4_f32)
#error "device toolchain lacks __builtin_amdgcn_wmma_f32_16x16x4_f32 (CDNA5 f32 WMMA)"
#endif

typedef float v2f __attribute__((ext_vector_type(2)));
typedef float v8f __attribute__((ext_vector_type(8)));

#define ACT_STRIDE 17     // 16x16 activation tile, +1 word to avoid bank conflicts
#define ODE_STRIDE 117    // odd -> per-lane row reads hit distinct banks; room for cols 0..115
#define RK4_STEPS  32

static __device__ __forceinline__ v8f wmma4(v2f a, v2f b, v8f c) {
  // D(16x16,f32) = A(16x4,f32) * B(4x16,f32) + C
  return __builtin_amdgcn_wmma_f32_16x16x4_f32(false, a, false, b, (short)0, c, false, false);
}

// B fragment: Bmat(k,n) = W[n*Kdim + k] (i.e. W^T), zero outside [Kdim,Nvalid).
// Branchless: clamped (always in-bounds) address, value select -> no EXEC branching.
static __device__ __forceinline__ v2f bfrag(const float* __restrict__ W, int Kdim, int Nvalid,
                                            int kb, int half, int n) {
  const int k0  = kb + 2 * half;
  const int nc  = (n      < Nvalid) ? n      : Nvalid - 1;
  const int kc0 = (k0     < Kdim)   ? k0     : Kdim - 1;
  const int kc1 = (k0 + 1 < Kdim)   ? k0 + 1 : Kdim - 1;
  const float w0 = W[nc * Kdim + kc0];
  const float w1 = W[nc * Kdim + kc1];
  v2f b;
  b.x = (n < Nvalid && k0     < Kdim) ? w0 : 0.0f;
  b.y = (n < Nvalid && k0 + 1 < Kdim) ? w1 : 0.0f;
  return b;
}

// A fragment from the 16x16 LDS activation tile (cols 10..15 are exact zeros)
static __device__ __forceinline__ v2f afrag_lds(const float* act, int kb, int half, int m) {
  const int k0 = kb + 2 * half;
  v2f a;
  a.x = act[m * ACT_STRIDE + k0];
  a.y = act[m * ACT_STRIDE + k0 + 1];
  return a;
}

// A fragment straight from x rows (coloff=10 -> params, coloff=0 -> inputs). Branchless.
static __device__ __forceinline__ v2f afrag_x(const float* __restrict__ x, int gbase, int coloff,
                                              int kb, int half, int m) {
  const int k0 = kb + 2 * half;
  const float* row = x + (long)(gbase + m) * 20;
  const float r0 = row[coloff + ((k0     < 10) ? k0     : 9)];
  const float r1 = row[coloff + ((k0 + 1 < 10) ? k0 + 1 : 9)];
  v2f a;
  a.x = (k0     < 10) ? r0 : 0.0f;
  a.y = (k0 + 1 < 10) ? r1 : 0.0f;
  return a;
}

static __device__ __forceinline__ void odefunc(const float* A, const float* y, float* k) {
#pragma unroll
  for (int i = 0; i < 10; ++i) {
    float t = 0.0f;
#pragma unroll
    for (int j = 0; j < 10; ++j) t = fmaf(A[i * 10 + j], y[j], t);
    k[i] = tanhf(t);
  }
}

__global__ __launch_bounds__(128, 1) void node_fused(
    const float* __restrict__ x,
    const float* __restrict__ fcIn_W,  const float* __restrict__ fcIn_b,
    const float* __restrict__ fcOut_W, const float* __restrict__ fcOut_b,
    const float* __restrict__ p0_W,    const float* __restrict__ p0_b,
    const float* __restrict__ p1_W,    const float* __restrict__ p1_b,
    const float* __restrict__ p2_W,    const float* __restrict__ p2_b,
    const float* __restrict__ pOut_W,  const float* __restrict__ pOut_b,
    float* __restrict__ out)
{
  __shared__ float s_ode[4 * 32 * ODE_STRIDE];      // per-wave: 32 samples x (A[100], y0)
  __shared__ float s_act[4][16 * ACT_STRIDE];       // per-wave 16x16 activation tile

  const int tid  = threadIdx.x;
  const int wave = tid >> 5;
  const int lane = tid & 31;
  const int half = lane >> 4;       // lane halves carry K split (ISA 7.12.2)
  const int m    = lane & 15;       // A: row index; B/C/D: column index

  const int blockBase = blockIdx.x * 128;
  float* act = s_act[wave];
  float* ode = &s_ode[wave * 32 * ODE_STRIDE];

  __builtin_prefetch(&x[(long)(blockBase + wave * 32 + lane) * 20], 0, 1);

  const v8f vzero = {0.f, 0.f, 0.f, 0.f, 0.f, 0.f, 0.f, 0.f};

  // ---- hoist all weight (B) fragments: block-invariant, built once ----
  v2f Bp0[3], Bp1[3], Bp2[3], BfcIn[3], BpOut[7][3];
#pragma unroll
  for (int c = 0; c < 3; ++c) {
    Bp0[c]   = bfrag(p0_W,   10, 10, 4 * c, half, m);
    Bp1[c]   = bfrag(p1_W,   10, 10, 4 * c, half, m);
    Bp2[c]   = bfrag(p2_W,   10, 10, 4 * c, half, m);
    BfcIn[c] = bfrag(fcIn_W, 10, 10, 4 * c, half, m);
#pragma unroll
    for (int t = 0; t < 7; ++t)
      BpOut[t][c] = bfrag(pOut_W, 10, 100, 4 * c, half, 16 * t + m);
  }
  const float bias0   = p0_b[(m < 10) ? m : 9] * ((m < 10) ? 1.0f : 0.0f);
  const float bias1   = p1_b[(m < 10) ? m : 9] * ((m < 10) ? 1.0f : 0.0f);
  const float bias2   = p2_b[(m < 10) ? m : 9] * ((m < 10) ? 1.0f : 0.0f);
  const float biasIn  = fcIn_b[(m < 10) ? m : 9];

#pragma unroll
  for (int tt = 0; tt < 2; ++tt) {
    const int gbase = blockBase + wave * 32 + tt * 16;  // first global sample of this tile
    const int lrow  = tt * 16;                          // first row in this wave's ode region

    // ---- hyper layer 0: H = relu(params @ p0_W^T + p0_b) ----
    v8f d = vzero;
#pragma unroll
    for (int c = 0; c < 3; ++c)
      d = wmma4(afrag_x(x, gbase, 10, 4 * c, half, m), Bp0[c], d);
#pragma unroll
    for (int r = 0; r < 8; ++r)
      act[(r + 8 * half) * ACT_STRIDE + m] = fmaxf(d[r] + bias0, 0.0f);

    // ---- hyper layer 1 ----
    {
      v2f a0 = afrag_lds(act, 0, half, m);
      v2f a1 = afrag_lds(act, 4, half, m);
      v2f a2 = afrag_lds(act, 8, half, m);
      v8f dd = vzero;
      dd = wmma4(a0, Bp1[0], dd);
      dd = wmma4(a1, Bp1[1], dd);
      dd = wmma4(a2, Bp1[2], dd);
      __builtin_amdgcn_wave_barrier();   // keep tile reads above the overwrite
#pragma unroll
      for (int r = 0; r < 8; ++r)
        act[(r + 8 * half) * ACT_STRIDE + m] = fmaxf(dd[r] + bias1, 0.0f);
    }

    // ---- hyper layer 2 ----
    {
      v2f a0 = afrag_lds(act, 0, half, m);
      v2f a1 = afrag_lds(act, 4, half, m);
      v2f a2 = afrag_lds(act, 8, half, m);
      v8f dd = vzero;
      dd = wmma4(a0, Bp2[0], dd);
      dd = wmma4(a1, Bp2[1], dd);
      dd = wmma4(a2, Bp2[2], dd);
      __builtin_amdgcn_wave_barrier();
#pragma unroll
      for (int r = 0; r < 8; ++r)
        act[(r + 8 * half) * ACT_STRIDE + m] = fmaxf(dd[r] + bias2, 0.0f);
    }

    // ---- output layer: per-sample A matrices (N=100, padded to 112 -> 7 tiles) ----
    // Unguarded stores: cols 100..111 are scratch, overwritten by the y0 stores below
    // (per-wave LDS ops are in-order), never read by phase 2.
    {
      v2f a0 = afrag_lds(act, 0, half, m);
      v2f a1 = afrag_lds(act, 4, half, m);
      v2f a2 = afrag_lds(act, 8, half, m);
#pragma unroll
      for (int t = 0; t < 7; ++t) {
        const int n = 16 * t + m;
        v8f dd = vzero;
        dd = wmma4(a0, BpOut[t][0], dd);
        dd = wmma4(a1, BpOut[t][1], dd);
        dd = wmma4(a2, BpOut[t][2], dd);
        const float bias = pOut_b[(n < 100) ? n : 99];
#pragma unroll
        for (int r = 0; r < 8; ++r)
          ode[(lrow + r + 8 * half) * ODE_STRIDE + n] = dd[r] + bias;
      }
    }

    // ---- fcIn: y0 = inputs @ fcIn_W^T + fcIn_b (cols 100..115; phase 2 reads 100..109) ----
    {
      v8f dd = vzero;
#pragma unroll
      for (int c = 0; c < 3; ++c)
        dd = wmma4(afrag_x(x, gbase, 0, 4 * c, half, m), BfcIn[c], dd);
#pragma unroll
      for (int r = 0; r < 8; ++r)
        ode[(lrow + r + 8 * half) * ODE_STRIDE + 100 + m] = dd[r] + biasIn;
    }
  }

  __builtin_amdgcn_wave_barrier();   // wave-private LDS; DS ops are in-order per wave

  // ---- phase 2: one sample per lane, fixed-step RK4 on dy/dt = tanh(A y) ----
  float Am[100], y[10];
  const float* row = &ode[lane * ODE_STRIDE];
#pragma unroll
  for (int i = 0; i < 100; ++i) Am[i] = row[i];
#pragma unroll
  for (int i = 0; i < 10; ++i)  y[i] = row[100 + i];

  const float h = 1.0f / RK4_STEPS;
  for (int s = 0; s < RK4_STEPS; ++s) {
    float k1[10], kt[10], yt[10], acc[10];
    odefunc(Am, y, k1);
#pragma unroll
    for (int i = 0; i < 10; ++i) { acc[i] = k1[i];                    yt[i] = fmaf(0.5f * h, k1[i], y[i]); }
    odefunc(Am, yt, kt);
#pragma unroll
    for (int i = 0; i < 10; ++i) { acc[i] = fmaf(2.f, kt[i], acc[i]); yt[i] = fmaf(0.5f * h, kt[i], y[i]); }
    odefunc(Am, yt, kt);
#pragma unroll
    for (int i = 0; i < 10; ++i) { acc[i] = fmaf(2.f, kt[i], acc[i]); yt[i] = fmaf(h, kt[i], y[i]); }
    odefunc(Am, yt, kt);
#pragma unroll
    for (int i = 0; i < 10; ++i) y[i] = fmaf(h * (1.f / 6.f), acc[i] + kt[i], y[i]);
  }

  // ---- fcOut ----
  const long gs = blockBase + wave * 32 + lane;
#pragma unroll
  for (int j = 0; j < 10; ++j) {
    float o = fcOut_b[j];
#pragma unroll
    for (int i = 0; i < 10; ++i) o = fmaf(fcOut_W[j * 10 + i], y[i], o);
    out[gs * 10 + j] = o;
  }
}

extern "C" void kernel_launch(void* const* d_in, const int* in_sizes, int n_in,
                              void* d_out, int out_size, void* d_ws, size_t ws_size,
                              hipStream_t stream) {
  const float* x       = (const float*)d_in[0];
  const float* fcIn_W  = (const float*)d_in[1];
  const float* fcIn_b  = (const float*)d_in[2];
  const float* fcOut_W = (const float*)d_in[3];
  const float* fcOut_b = (const float*)d_in[4];
  const float* p0_W    = (const float*)d_in[5];
  const float* p0_b    = (const float*)d_in[6];
  const float* p1_W    = (const float*)d_in[7];
  const float* p1_b    = (const float*)d_in[8];
  const float* p2_W    = (const float*)d_in[9];
  const float* p2_b    = (const float*)d_in[10];
  const float* pOut_W  = (const float*)d_in[11];
  const float* pOut_b  = (const float*)d_in[12];
  float* out = (float*)d_out;

  const int B = in_sizes[0] / 20;            // 131072; divisible by 128
  dim3 grid(B / 128), block(128);
  hipLaunchKernelGGL(node_fused, grid, block, 0, stream,
                     x, fcIn_W, fcIn_b, fcOut_W, fcOut_b,
                     p0_W, p0_b, p1_W, p1_b, p2_W, p2_b, pOut_W, pOut_b, out);
}